// HybridContrastiveLoss_9637906612331
// MI455X (gfx1250) — compile-verified
//
#include <hip/hip_runtime.h>
#include <hip/hip_bf16.h>
#include <math.h>

// ---------------------------------------------------------------------------
// MoCo-style hybrid contrastive loss, forward only.
//   loss = mean_n [ logsumexp([pos_n, online_n . queue_:,k] / t) - pos_n / t ]
// (alpha * L + (1-alpha) * L == L in the forward pass.)
//
// GEMM (1024x256 @ 256x65536) fused with streaming logsumexp via
// V_WMMA_F32_16X16X4_F32.  Memory-bound: ~66 MB HBM -> ~2.8us floor at
// 23.3 TB/s.  NT=4 deferred softmax update cuts v_exp_f32 count from
// 2/elem to 1.25/elem so transcendentals stay under the memory floor.
// ---------------------------------------------------------------------------

typedef __attribute__((ext_vector_type(2))) float v2f;
typedef __attribute__((ext_vector_type(8))) float v8f;

#define NROWS   1024
#define DDIM    256
#define KDIM    65536
#define KSPLIT  8                       // K strips (grid.y)
#define WAVES   8                       // waves per workgroup (wave32)
#define TPB     (WAVES * 32)
#define KSTRIP  (KDIM / KSPLIT)         // 8192 cols per workgroup
#define COLS_PER_WAVE (KSTRIP / WAVES)  // 1024 cols per wave
#define TILES_PER_WAVE (COLS_PER_WAVE / 16) // 64 WMMA column tiles per wave
#define NT      4                       // column tiles per softmax update
#define NGROUPS (TILES_PER_WAVE / NT)   // 16 groups per wave

// ---------------------------------------------------------------------------
// Kernel 1: per (16-row tile, K strip): GEMM tile groups + deferred online
// logsumexp.  Emits per-(row, strip) partial (m, s) into d_ws.
// ---------------------------------------------------------------------------
__global__ __launch_bounds__(TPB)
void moco_lse_kernel(const float* __restrict__ online,
                     const float* __restrict__ queue,
                     const float* __restrict__ temp,
                     float2* __restrict__ partials)
{
    __shared__ float ldsA[16 * DDIM];   // 16 KB: A tile, row-major [16][256]
    __shared__ float ldsM[16 * 128];    // 8 KB: per-row per-(wave,col) max
    __shared__ float ldsS[16 * 128];    // 8 KB: per-row per-(wave,col) sumexp

    const int tid  = threadIdx.x;
    const int wave = tid >> 5;
    const int lane = tid & 31;
    const int row0 = blockIdx.x * 16;

    const float invT = 1.0f / temp[0];  // uniform scalar load

    // ---- stage A tile (16 rows x 256) into LDS: each thread copies 16 floats
    {
        const int r = tid >> 4;                 // 0..15
        const int d = (tid & 15) * 16;          // 0..240
        const float4* src = (const float4*)(online + (size_t)(row0 + r) * DDIM + d);
        float4* dst = (float4*)(&ldsA[r * DDIM + d]);
        dst[0] = src[0]; dst[1] = src[1]; dst[2] = src[2]; dst[3] = src[3];
    }
    __syncthreads();

    const int laneHalf = lane >> 4;   // 0: lanes 0-15, 1: lanes 16-31
    const int lane16   = lane & 15;
    const int aoff     = laneHalf * 2; // A frag: low half k={0,1}, high half k={2,3}
    const int boff     = laneHalf * 2; // B frag: low half rows {0,1}, high {2,3}

    // Streaming logsumexp state: each lane holds 8 C-rows (M = i + laneHalf*8).
    float m[8], s[8];
#pragma unroll
    for (int i = 0; i < 8; ++i) { m[i] = -INFINITY; s[i] = 0.0f; }

    // Scalar column base of this wave's strip (lane16 added for B/C columns).
    const size_t cg0 = (size_t)blockIdx.y * KSTRIP + (size_t)wave * COLS_PER_WAVE;

    for (int g = 0; g < NGROUPS; ++g) {
        const float* bcol = queue + cg0 + (size_t)g * (NT * 16) + lane16;

        // Wide prefetch of the NEXT group's B region (256 rows x 256B):
        // per-lane addresses cover 16 rows x 2 cachelines per instruction.
        if (g + 1 < NGROUPS) {
            const float* nb = queue + cg0 + (size_t)(g + 1) * (NT * 16);
            const int pr = lane >> 1;            // 0..15: row within 16-row band
            const int pc = (lane & 1) * 32;      // 0 / 128B within the 256B span
#pragma unroll
            for (int p = 0; p < 16; ++p) {
                __builtin_prefetch(nb + (size_t)(p * 16 + pr) * KDIM + pc, 0, 1);
            }
        }

        v8f acc0 = {}, acc1 = {}, acc2 = {}, acc3 = {};
#pragma unroll 4
        for (int d4 = 0; d4 < DDIM / 4; ++d4) {
            // A fragment (16x4 fp32): lane row M = lane16, k = 4*d4 + aoff + {0,1}
            const float2 af = *(const float2*)(&ldsA[lane16 * DDIM + 4 * d4 + aoff]);
            v2f a; a.x = af.x; a.y = af.y;

            // B fragments for 4 adjacent 16-col tiles (rows 4*d4+boff+{0,1}).
            const float* brow0 = bcol + (size_t)(4 * d4 + boff) * KDIM;
            const float* brow1 = brow0 + KDIM;
            v2f b0; b0.x = brow0[0];  b0.y = brow1[0];
            v2f b1; b1.x = brow0[16]; b1.y = brow1[16];
            v2f b2; b2.x = brow0[32]; b2.y = brow1[32];
            v2f b3; b3.x = brow0[48]; b3.y = brow1[48];

            acc0 = __builtin_amdgcn_wmma_f32_16x16x4_f32(false, a, false, b0, (short)0, acc0, false, false);
            acc1 = __builtin_amdgcn_wmma_f32_16x16x4_f32(false, a, false, b1, (short)0, acc1, false, false);
            acc2 = __builtin_amdgcn_wmma_f32_16x16x4_f32(false, a, false, b2, (short)0, acc2, false, false);
            acc3 = __builtin_amdgcn_wmma_f32_16x16x4_f32(false, a, false, b3, (short)0, acc3, false, false);
        }

        // Deferred online-softmax update: per row, 4 new values (4 columns)
        // -> 1 rescale exp + 4 element exps (1.25 exp/element).
#pragma unroll
        for (int i = 0; i < 8; ++i) {
            const float x0 = acc0[i] * invT;
            const float x1 = acc1[i] * invT;
            const float x2 = acc2[i] * invT;
            const float x3 = acc3[i] * invT;
            const float xm = fmaxf(fmaxf(x0, x1), fmaxf(x2, x3));
            const float mn = fmaxf(m[i], xm);
            const float sc = __expf(m[i] - mn);
            s[i] = s[i] * sc + ((__expf(x0 - mn) + __expf(x1 - mn)) +
                                (__expf(x2 - mn) + __expf(x3 - mn)));
            m[i] = mn;
        }
    }

    // ---- combine across lanes + waves through LDS
    {
        const int cslot = wave * 16 + lane16;   // 0..127
#pragma unroll
        for (int i = 0; i < 8; ++i) {
            const int r = i + laneHalf * 8;     // C row within the 16-row tile
            ldsM[r * 128 + cslot] = m[i];
            ldsS[r * 128 + cslot] = s[i];
        }
    }
    __syncthreads();

    if (tid < 16) {
        const int r = tid;
        float mr = -INFINITY;
        for (int c = 0; c < 128; ++c) mr = fmaxf(mr, ldsM[r * 128 + c]);
        float sr = 0.0f;
        for (int c = 0; c < 128; ++c) sr += ldsS[r * 128 + c] * __expf(ldsM[r * 128 + c] - mr);
        partials[(size_t)(row0 + r) * KSPLIT + blockIdx.y] = make_float2(mr, sr);
    }
}

// ---------------------------------------------------------------------------
// Kernel 2: one 1024-thread block.  Per row: pos dot, fold KSPLIT partials +
// positive logit, lse - pos/t; tree-reduce; mean -> d_out[0].
// ---------------------------------------------------------------------------
__global__ __launch_bounds__(1024)
void moco_finish_kernel(const float* __restrict__ online,
                        const float* __restrict__ mom,
                        const float* __restrict__ temp,
                        const float2* __restrict__ partials,
                        float* __restrict__ out)
{
    __shared__ float red[1024];
    const int n = threadIdx.x;
    const float invT = 1.0f / temp[0];

    // pos_n = dot(online[n], mom[n]) over D=256
    const float4* a4 = (const float4*)(online + (size_t)n * DDIM);
    const float4* b4 = (const float4*)(mom    + (size_t)n * DDIM);
    float dot = 0.0f;
#pragma unroll 4
    for (int i = 0; i < DDIM / 4; ++i) {
        const float4 a = a4[i], b = b4[i];
        dot += a.x * b.x + a.y * b.y + a.z * b.z + a.w * b.w;
    }
    const float posl = dot * invT;

    // start with the positive logit, fold the strip partials
    float m = posl, s = 1.0f;
    for (int p = 0; p < KSPLIT; ++p) {
        const float2 ps = partials[(size_t)n * KSPLIT + p];
        const float mn = fmaxf(m, ps.x);
        s = s * __expf(m - mn) + ps.y * __expf(ps.x - mn);
        m = mn;
    }
    red[n] = (m + __logf(s)) - posl;     // logsumexp - pos/t
    __syncthreads();

    for (int stride = 512; stride > 0; stride >>= 1) {
        if (n < stride) red[n] += red[n + stride];
        __syncthreads();
    }
    if (n == 0) out[0] = red[0] * (1.0f / (float)NROWS);
}

// ---------------------------------------------------------------------------
extern "C" void kernel_launch(void* const* d_in, const int* in_sizes, int n_in,
                              void* d_out, int out_size, void* d_ws, size_t ws_size,
                              hipStream_t stream)
{
    const float* online = (const float*)d_in[0];   // [1024, 256]
    const float* mom    = (const float*)d_in[1];   // [1024, 256]
    const float* queue  = (const float*)d_in[2];   // [256, 65536]
    const float* temp   = (const float*)d_in[3];   // [1]
    float* out = (float*)d_out;

    float2* partials = (float2*)d_ws;              // 1024 * KSPLIT * 8B = 64 KB

    dim3 grid1(NROWS / 16, KSPLIT);                // 64 x 8 workgroups
    moco_lse_kernel<<<grid1, TPB, 0, stream>>>(online, queue, temp, partials);

    moco_finish_kernel<<<1, 1024, 0, stream>>>(online, mom, temp, partials, out);
}